// LocalState_9053791060532
// MI455X (gfx1250) — compile-verified
//
#include <hip/hip_runtime.h>
#include <math.h>

// ---------------------------------------------------------------------------
// LocalState attention for MI455X (gfx1250, wave32, WMMA f32 16x16x4)
//
// B=2, C=512, T=2048, HEADS=8, HD=64, NFREQS=NDECAY=4.
// Freq bias folded into an extended head dim (64+8=72) via the cos-difference
// identity; decay bias collapsed to -|t-s| * dcoef[s]; attention computed
// flash-style (no TxT materialization; everything stays L2-resident).
// GEMMs are 2x2 register-blocked (32x32/wave) -> 2 L2 loads per WMMA.
// ---------------------------------------------------------------------------

#define B_      2
#define C_      512
#define T_      2048
#define HEADS_  8
#define HD_     64
#define NF_     4
#define ND_     4
#define KD_     72            // 64 + 2*NF extended head dim (multiple of 4)

typedef float v2f __attribute__((ext_vector_type(2)));
typedef float v8f __attribute__((ext_vector_type(8)));

static_assert(KD_ % 4 == 0, "KD must be multiple of WMMA K=4");

__device__ __forceinline__ v8f wmma_k4(v2f a, v2f b, v8f c) {
    // V_WMMA_F32_16X16X4_F32 : D = A(16x4) * B(4x16) + C(16x16), f32 exact
    return __builtin_amdgcn_wmma_f32_16x16x4_f32(false, a, false, b,
                                                 (short)0, c, false, false);
}

// ---------------------------------------------------------------------------
// Generic GEMM:  Y[b] = W(MxK) * X[b](KxT) + bias (+ resid)
// One wave per 32x32 output block (2x2 WMMA tiles, 4 accumulators).
// W row-major; X/Y channel-major [rows,T].
// A-operand tile mt: lane m=lane&15 reads W[m0+16*mt+m][k+2*grp+{0,1}]
// B-operand tile nt: lane n reads X[k+2*grp+{0,1}][t0+16*nt+n] (coalesced)
// ---------------------------------------------------------------------------
__global__ __launch_bounds__(32) void gemm_ws(
    const float* __restrict__ W, const float* __restrict__ bias,
    const float* __restrict__ X, float* __restrict__ Y,
    const float* __restrict__ resid, int M, int K)
{
    const int t0   = blockIdx.x * 32;
    const int m0   = blockIdx.y * 32;
    const int b    = blockIdx.z;
    const int lane = threadIdx.x & 31;
    const int n    = lane & 15;
    const int grp  = lane >> 4;

    const float* Xb = X + (size_t)b * K * T_;
    v8f acc00 = {}, acc01 = {}, acc10 = {}, acc11 = {};
    for (int k = 0; k < K; k += 4) {
        const int ka = k + 2 * grp;
        v2f a0, a1, b0, b1;
        a0.x = W[(size_t)(m0 + n)      * K + ka];
        a0.y = W[(size_t)(m0 + n)      * K + ka + 1];
        a1.x = W[(size_t)(m0 + 16 + n) * K + ka];
        a1.y = W[(size_t)(m0 + 16 + n) * K + ka + 1];
        b0.x = Xb[(size_t)ka       * T_ + t0 + n];
        b0.y = Xb[(size_t)(ka + 1) * T_ + t0 + n];
        b1.x = Xb[(size_t)ka       * T_ + t0 + 16 + n];
        b1.y = Xb[(size_t)(ka + 1) * T_ + t0 + 16 + n];
        acc00 = wmma_k4(a0, b0, acc00);
        acc01 = wmma_k4(a0, b1, acc01);
        acc10 = wmma_k4(a1, b0, acc10);
        acc11 = wmma_k4(a1, b1, acc11);
    }

    float*       Yb = Y + (size_t)b * M * T_;
    const float* Rb = resid ? resid + (size_t)b * M * T_ : nullptr;
    #pragma unroll
    for (int r = 0; r < 8; ++r) {
        const int m_lo = m0 + grp * 8 + r;            // D: VGPR r -> row r+8*grp
        const int m_hi = m_lo + 16;
        const float bl = bias[m_lo], bh = bias[m_hi];
        const size_t o00 = (size_t)m_lo * T_ + t0 + n;
        const size_t o01 = o00 + 16;
        const size_t o10 = (size_t)m_hi * T_ + t0 + n;
        const size_t o11 = o10 + 16;
        float v00 = acc00[r] + bl, v01 = acc01[r] + bl;
        float v10 = acc10[r] + bh, v11 = acc11[r] + bh;
        if (Rb) { v00 += Rb[o00]; v01 += Rb[o01]; v10 += Rb[o10]; v11 += Rb[o11]; }
        Yb[o00] = v00; Yb[o01] = v01; Yb[o10] = v10; Yb[o11] = v11;
    }
}

// ---------------------------------------------------------------------------
// Prep: build extended Qe/Ke [B,H,72,T], transposed content [B,H,T,64],
// and decay coefficient dcoef[B,H,T]. One thread per (b,h,t).
//   Qe rows 0..63  = Yq/8 (score scale folded in);  Ke rows 0..63 = Yk
//   Qe rows 64+2f  = cos(2πt/(f+1)) * Yqf/2,  65+2f = sin(...) * Yqf/2
//   Ke rows 64+2f  = cos(2πt/(f+1)),          65+2f = sin(...)
//   dcoef = Σ_f (f+1)*sigmoid(Yqd)/4
// ---------------------------------------------------------------------------
__global__ __launch_bounds__(256) void prep_kernel(
    const float* __restrict__ Yq, const float* __restrict__ Yk,
    const float* __restrict__ Yc, const float* __restrict__ Yqf,
    const float* __restrict__ Yqd,
    float* __restrict__ Qe, float* __restrict__ Ke,
    float* __restrict__ ContT, float* __restrict__ dcoef)
{
    const int idx = blockIdx.x * blockDim.x + threadIdx.x;
    if (idx >= B_ * HEADS_ * T_) return;
    const int t = idx % T_;
    const int h = (idx / T_) % HEADS_;
    const int b = idx / (T_ * HEADS_);

    const float* yq  = Yq  + (size_t)b * C_ * T_;
    const float* yk  = Yk  + (size_t)b * C_ * T_;
    const float* yc  = Yc  + (size_t)b * C_ * T_;
    const float* yqf = Yqf + (size_t)b * (HEADS_ * NF_) * T_;
    const float* yqd = Yqd + (size_t)b * (HEADS_ * ND_) * T_;

    const size_t eb = (size_t)(b * HEADS_ + h) * KD_ * T_;
    const size_t cb = ((size_t)(b * HEADS_ + h) * T_ + t) * HD_;

    for (int c = 0; c < HD_; ++c) {
        const size_t src = (size_t)(h * HD_ + c) * T_ + t;
        Qe[eb + (size_t)c * T_ + t] = yq[src] * 0.125f;   // 1/sqrt(64)
        Ke[eb + (size_t)c * T_ + t] = yk[src];
        ContT[cb + c]               = yc[src];
    }

    const float twopi = 6.28318530717958647692f;
    float dc = 0.0f;
    #pragma unroll
    for (int f = 0; f < NF_; ++f) {
        const float ang = twopi * (float)t / (float)(f + 1);
        const float cs = cosf(ang), sn = sinf(ang);
        const float fq = yqf[(size_t)(h * NF_ + f) * T_ + t] * 0.5f; // 1/sqrt(4)
        Qe[eb + (size_t)(HD_ + 2 * f)     * T_ + t] = cs * fq;
        Qe[eb + (size_t)(HD_ + 2 * f + 1) * T_ + t] = sn * fq;
        Ke[eb + (size_t)(HD_ + 2 * f)     * T_ + t] = cs;
        Ke[eb + (size_t)(HD_ + 2 * f + 1) * T_ + t] = sn;
        const float qd  = yqd[(size_t)(h * ND_ + f) * T_ + t];
        dc += (float)(f + 1) / (1.0f + __expf(-qd));
    }
    dcoef[(size_t)(b * HEADS_ + h) * T_ + t] = dc * 0.25f; // /sqrt(4) * (sig/2)
}

// ---------------------------------------------------------------------------
// Fused flash attention. 4 waves/block; each wave owns one 16-column s-tile of
// one (b,h). Loops over 128 t-tiles: 18 score WMMAs (K=72), online softmax
// (per-lane m/l scalars + one shfl_xor(16) cross-half reduce), P staged via
// 1KB LDS/wave to build B-operands, 16 AV WMMAs (4 c-tiles x 4 k-steps).
// ---------------------------------------------------------------------------
__global__ __launch_bounds__(128) void attn_kernel(
    const float* __restrict__ Qe, const float* __restrict__ Ke,
    const float* __restrict__ ContT, const float* __restrict__ dcoef,
    float* __restrict__ Res)
{
    __shared__ float lds_p[4][16 * 16];

    const int wib  = threadIdx.x >> 5;
    const int lane = threadIdx.x & 31;
    const int n    = lane & 15;
    const int grp  = lane >> 4;

    const int wid    = blockIdx.x * 4 + wib;
    const int ntile  = T_ / 16;                    // 128
    const int s_tile = wid % ntile;
    const int h      = (wid / ntile) % HEADS_;
    const int b      = wid / (ntile * HEADS_);
    const int s0     = s_tile * 16;

    const size_t eb = (size_t)(b * HEADS_ + h) * KD_ * T_;
    const size_t cb = (size_t)(b * HEADS_ + h) * T_ * HD_;

    // Qe B-operands held in registers for all 18 K-steps (36 VGPRs)
    v2f qb[KD_ / 4];
    #pragma unroll
    for (int kk = 0; kk < KD_ / 4; ++kk) {
        const int c = 4 * kk + 2 * grp;
        qb[kk].x = Qe[eb + (size_t)c       * T_ + s0 + n];
        qb[kk].y = Qe[eb + (size_t)(c + 1) * T_ + s0 + n];
    }
    const float dc = dcoef[(size_t)(b * HEADS_ + h) * T_ + s0 + n];

    float mrun = -1e30f, lrun = 0.0f;
    v8f o0 = {}, o1 = {}, o2 = {}, o3 = {};
    float* myp = lds_p[wib];

    for (int t0 = 0; t0 < T_; t0 += 16) {
        // ---- scores S = Ke^T * Qe over extended dim 72 ----
        v8f sacc = {};
        #pragma unroll
        for (int kk = 0; kk < KD_ / 4; ++kk) {
            const int c = 4 * kk + 2 * grp;
            v2f a;
            a.x = Ke[eb + (size_t)c       * T_ + t0 + n];
            a.y = Ke[eb + (size_t)(c + 1) * T_ + t0 + n];
            sacc = wmma_k4(a, qb[kk], sacc);
        }

        // ---- decay bias + diagonal mask; tile max ----
        float sv[8];
        float tm = -1e30f;
        #pragma unroll
        for (int r = 0; r < 8; ++r) {
            const int t = t0 + grp * 8 + r;        // D row mapping
            const int s = s0 + n;
            float v = sacc[r] - fabsf((float)(t - s)) * dc;
            if (t == s) v = -100.0f;
            sv[r] = v;
            tm = fmaxf(tm, v);
        }
        tm = fmaxf(tm, __shfl_xor(tm, 16, 32));    // combine half-wave rows
        const float nm    = fmaxf(mrun, tm);
        const float alpha = __expf(mrun - nm);

        float rs = 0.0f;
        #pragma unroll
        for (int r = 0; r < 8; ++r) {
            const float p = __expf(sv[r] - nm);
            myp[(grp * 8 + r) * 16 + n] = p;       // stage P in LDS
            rs += p;
        }
        rs  += __shfl_xor(rs, 16, 32);
        lrun = lrun * alpha + rs;
        mrun = nm;

        #pragma unroll
        for (int r = 0; r < 8; ++r) {
            o0[r] *= alpha; o1[r] *= alpha; o2[r] *= alpha; o3[r] *= alpha;
        }
        __syncthreads();                           // LDS store -> load

        // ---- O += content * P  (K dim = 16 local t, 4 steps) ----
        #pragma unroll
        for (int kk = 0; kk < 4; ++kk) {
            const int tl = 4 * kk + 2 * grp;
            v2f bp;
            bp.x = myp[tl * 16 + n];
            bp.y = myp[(tl + 1) * 16 + n];
            const size_t r0 = cb + (size_t)(t0 + tl)     * HD_;
            const size_t r1 = cb + (size_t)(t0 + tl + 1) * HD_;
            v2f a0, a1, a2, a3;
            a0.x = ContT[r0 + 0  + n]; a0.y = ContT[r1 + 0  + n];
            a1.x = ContT[r0 + 16 + n]; a1.y = ContT[r1 + 16 + n];
            a2.x = ContT[r0 + 32 + n]; a2.y = ContT[r1 + 32 + n];
            a3.x = ContT[r0 + 48 + n]; a3.y = ContT[r1 + 48 + n];
            o0 = wmma_k4(a0, bp, o0);
            o1 = wmma_k4(a1, bp, o1);
            o2 = wmma_k4(a2, bp, o2);
            o3 = wmma_k4(a3, bp, o3);
        }
        __syncthreads();                           // protect LDS reuse
    }

    // ---- normalize and write result in [B,C,T] layout for final GEMM ----
    const float rl   = 1.0f / lrun;
    float*     resb  = Res + (size_t)b * C_ * T_;
    const int  col   = s0 + n;
    #pragma unroll
    for (int r = 0; r < 8; ++r) {
        const int crow = grp * 8 + r;
        resb[(size_t)(h * HD_ + 0  + crow) * T_ + col] = o0[r] * rl;
        resb[(size_t)(h * HD_ + 16 + crow) * T_ + col] = o1[r] * rl;
        resb[(size_t)(h * HD_ + 32 + crow) * T_ + col] = o2[r] * rl;
        resb[(size_t)(h * HD_ + 48 + crow) * T_ + col] = o3[r] * rl;
    }
}

// ---------------------------------------------------------------------------
extern "C" void kernel_launch(void* const* d_in, const int* in_sizes, int n_in,
                              void* d_out, int out_size, void* d_ws, size_t ws_size,
                              hipStream_t stream) {
    (void)in_sizes; (void)n_in; (void)out_size; (void)ws_size;
    const float* x   = (const float*)d_in[0];
    const float* Wq  = (const float*)d_in[1];
    const float* bq  = (const float*)d_in[2];
    const float* Wk  = (const float*)d_in[3];
    const float* bk  = (const float*)d_in[4];
    const float* Wc  = (const float*)d_in[5];
    const float* bc  = (const float*)d_in[6];
    const float* Wqf = (const float*)d_in[7];
    const float* bqf = (const float*)d_in[8];
    const float* Wqd = (const float*)d_in[9];
    const float* bqd = (const float*)d_in[10];
    const float* Wp  = (const float*)d_in[11];
    const float* bp  = (const float*)d_in[12];
    float*       out = (float*)d_out;

    // workspace carve-up (floats)
    float* ws = (float*)d_ws;
    size_t off = 0;
    float* Yq    = ws + off; off += (size_t)B_ * C_ * T_;              // 2.10M
    float* Yk    = ws + off; off += (size_t)B_ * C_ * T_;
    float* Yc    = ws + off; off += (size_t)B_ * C_ * T_;
    float* Yqf   = ws + off; off += (size_t)B_ * HEADS_ * NF_ * T_;
    float* Yqd   = ws + off; off += (size_t)B_ * HEADS_ * ND_ * T_;
    float* Qe    = ws + off; off += (size_t)B_ * HEADS_ * KD_ * T_;
    float* Ke    = ws + off; off += (size_t)B_ * HEADS_ * KD_ * T_;
    float* ContT = ws + off; off += (size_t)B_ * HEADS_ * T_ * HD_;
    float* dcoef = ws + off; off += (size_t)B_ * HEADS_ * T_;
    float* Res   = Yq;  // Yq dead after prep; reuse for attention output

    const dim3 blk32(32), blk128(128), blk256(256);
    const dim3 gC (T_ / 32, C_ / 32, B_);                 // 512-row GEMMs, 32x32/wave
    const dim3 gF (T_ / 32, (HEADS_ * NF_) / 32, B_);     // 32-row GEMMs

    gemm_ws<<<gC, blk32, 0, stream>>>(Wq,  bq,  x, Yq,  nullptr, C_,            C_);
    gemm_ws<<<gC, blk32, 0, stream>>>(Wk,  bk,  x, Yk,  nullptr, C_,            C_);
    gemm_ws<<<gC, blk32, 0, stream>>>(Wc,  bc,  x, Yc,  nullptr, C_,            C_);
    gemm_ws<<<gF, blk32, 0, stream>>>(Wqf, bqf, x, Yqf, nullptr, HEADS_ * NF_,  C_);
    gemm_ws<<<gF, blk32, 0, stream>>>(Wqd, bqd, x, Yqd, nullptr, HEADS_ * ND_,  C_);

    const int nprep = B_ * HEADS_ * T_;
    prep_kernel<<<(nprep + 255) / 256, blk256, 0, stream>>>(
        Yq, Yk, Yc, Yqf, Yqd, Qe, Ke, ContT, dcoef);

    const int nwaves = B_ * HEADS_ * (T_ / 16);
    attn_kernel<<<nwaves / 4, blk128, 0, stream>>>(Qe, Ke, ContT, dcoef, Res);

    gemm_ws<<<gC, blk32, 0, stream>>>(Wp, bp, Res, out, x, C_, C_);   // +residual
}